// InPlaceTransform_28810640621832
// MI455X (gfx1250) — compile-verified
//
#include <hip/hip_runtime.h>
#include <math.h>

// Problem constants (match reference)
#define B_DIM 32768
#define D_DIM 512
#define K_BIN 10
#define PSTR  33              // floats per feature in the knot table: cw[11] | ch[11] | d[11]
#define TAILB 10.0f
#define MBW   1e-3f
#define MBH   1e-3f
#define MDER  1e-3f

#define ROWS_PER_BLOCK 8
#define THREADS 256

typedef unsigned int u32x4 __attribute__((ext_vector_type(4)));
typedef int          i32x8 __attribute__((ext_vector_type(8)));
typedef int          i32x4 __attribute__((ext_vector_type(4)));

__device__ __forceinline__ float softplusf(float v) {
    return v > 20.f ? v : log1pf(expf(v));
}

// ---------------------------------------------------------------------------
// Kernel 1: build the per-feature knot table P[d][33] = { cw[0..10], ch[0..10], dv[0..10] }
// One thread per feature (D=512). Tiny kernel; runs once before the main pass.
// ---------------------------------------------------------------------------
__global__ void __launch_bounds__(256)
rqs_prep_kernel(const float* __restrict__ uw,
                const float* __restrict__ uh,
                const float* __restrict__ ud,
                float* __restrict__ ptab) {
    const int d = blockIdx.x * blockDim.x + threadIdx.x;
    if (d >= D_DIM) return;
    float* p = ptab + d * PSTR;

    // widths -> cumulative x-knots cw
    {
        const float* r = uw + d * K_BIN;
        float m = r[0];
        #pragma unroll
        for (int k = 1; k < K_BIN; ++k) m = fmaxf(m, r[k]);
        float e[K_BIN]; float s = 0.f;
        #pragma unroll
        for (int k = 0; k < K_BIN; ++k) { e[k] = expf(r[k] - m); s += e[k]; }
        const float inv = 1.f / s;
        float cum = 0.f;
        p[0] = -TAILB;
        #pragma unroll
        for (int k = 0; k < K_BIN; ++k) {
            cum += MBW + (1.f - MBW * (float)K_BIN) * (e[k] * inv);
            p[1 + k] = (k == K_BIN - 1) ? TAILB : fmaf(2.f * TAILB, cum, -TAILB);
        }
    }
    // heights -> cumulative y-knots ch
    {
        const float* r = uh + d * K_BIN;
        float m = r[0];
        #pragma unroll
        for (int k = 1; k < K_BIN; ++k) m = fmaxf(m, r[k]);
        float e[K_BIN]; float s = 0.f;
        #pragma unroll
        for (int k = 0; k < K_BIN; ++k) { e[k] = expf(r[k] - m); s += e[k]; }
        const float inv = 1.f / s;
        float cum = 0.f;
        p[11] = -TAILB;
        #pragma unroll
        for (int k = 0; k < K_BIN; ++k) {
            cum += MBH + (1.f - MBH * (float)K_BIN) * (e[k] * inv);
            p[12 + k] = (k == K_BIN - 1) ? TAILB : fmaf(2.f * TAILB, cum, -TAILB);
        }
    }
    // derivatives: boundary pad is softplus(softplus^-1(1-MDER)) + MDER == 1.0 exactly
    {
        const float* r = ud + d * (K_BIN - 1);
        p[22] = 1.0f;
        #pragma unroll
        for (int k = 1; k < K_BIN; ++k) p[22 + k] = MDER + softplusf(r[k - 1]);
        p[32] = 1.0f;
    }
}

// ---------------------------------------------------------------------------
// Per-element spline evaluation against LDS-resident knots for one feature.
// p points at the 33-float feature record in LDS.
// ---------------------------------------------------------------------------
__device__ __forceinline__ float rqs_eval(float xv, const float* __restrict__ p, float& acc) {
    const bool inside = (xv >= -TAILB) && (xv <= TAILB);
    const float xc = fminf(fmaxf(xv, -TAILB), TAILB);

    // bin search over interior knots cw[1..K-1] (EPS-bumped last edge never fires for clipped x)
    int idx = 0;
    #pragma unroll
    for (int j = 1; j < K_BIN; ++j) idx += (xc >= p[j]) ? 1 : 0;

    const float cwl = p[idx];
    const float w   = p[idx + 1] - cwl;
    const float chl = p[11 + idx];
    const float h   = p[12 + idx] - chl;
    const float dl  = p[22 + idx];
    const float dr  = p[23 + idx];

    const float delta = h / w;
    const float th    = (xc - cwl) / w;
    const float omt   = 1.f - th;
    const float t1m   = th * omt;

    const float num = h * (delta * th * th + dl * t1m);
    const float den = delta + (dl + dr - 2.f * delta) * t1m;
    const float yv  = chl + num / den;

    const float dnum = delta * delta * (dr * th * th + 2.f * delta * t1m + dl * omt * omt);
    const float lad  = logf(dnum) - 2.f * logf(den);

    acc += inside ? lad : 0.f;
    return inside ? yv : xv;
}

// ---------------------------------------------------------------------------
// Kernel 2: main streaming pass. Block = 256 threads, handles 8 rows x 512 cols.
// Stage the 66KB knot table into LDS once per block via the Tensor Data Mover,
// then stream x with float2 (b64) loads/stores. Deterministic per-row reduction.
// ---------------------------------------------------------------------------
__global__ void __launch_bounds__(THREADS)
rqs_main_kernel(const float* __restrict__ x,
                const float* __restrict__ ptab,
                float* __restrict__ y,
                float* __restrict__ logdet) {
    __shared__ float lds_raw[D_DIM * PSTR + 16];   // 66 KB knot table + reduction scratch

    // ---- TDM: DMA the knot table (16896 f32) from global into LDS offset 0 ----
    if (threadIdx.x < 32) {   // one wave issues; TDM ignores EXEC, tracked by TENSORcnt
        const unsigned long long ga = (unsigned long long)(uintptr_t)ptab;
        const unsigned int nelem = (unsigned int)(D_DIM * PSTR);   // 16896, fits tile_dim0[15:0]

        u32x4 g0;
        g0[0] = 1u;                                            // count=1, user descriptor
        g0[1] = 0u;                                            // lds_addr (bytes) = 0
        g0[2] = (unsigned int)ga;                              // global_addr[31:0]
        g0[3] = ((unsigned int)(ga >> 32) & 0x01FFFFFFu)       // global_addr[56:32]
              | 0x80000000u;                                   // type=2 ("image")

        i32x8 g1;
        g1[0] = (int)(2u << 16);                               // data_size=2 (4 bytes), mask=0
        g1[1] = (int)((nelem & 0xFFFFu) << 16);                // tensor_dim0[15:0]
        g1[2] = (int)((nelem >> 16) | (1u << 16));             // tensor_dim0[31:16] | tensor_dim1=1
        g1[3] = (int)((nelem & 0xFFFFu) << 16);                // tile_dim0 = 16896
        g1[4] = 1;                                             // tile_dim1 = 1, tile_dim2 = 0
        g1[5] = (int)nelem;                                    // tensor_dim0_stride[31:0]
        g1[6] = (int)((nelem & 0xFFFFu) << 16);                // stride0[47:32]=0 | stride1[15:0]
        g1[7] = 0;                                             // tensor_dim1_stride[47:16]

        i32x4 z4; z4[0] = 0; z4[1] = 0; z4[2] = 0; z4[3] = 0;  // groups 2/3 unused (<=2D)
        i32x8 z8;
        #pragma unroll
        for (int i = 0; i < 8; ++i) z8[i] = 0;
        __builtin_amdgcn_tensor_load_to_lds(g0, g1, z4, z4, z8, 0);
    }
    __builtin_amdgcn_s_wait_tensorcnt(0);
    __syncthreads();

    const int tid = threadIdx.x;
    const int c0  = tid * 2;                                   // two adjacent features/thread
    const float* __restrict__ pA = &lds_raw[c0 * PSTR];
    const float* __restrict__ pB = &lds_raw[(c0 + 1) * PSTR];
    float* redbuf = &lds_raw[D_DIM * PSTR];

    const long long row0 = (long long)blockIdx.x * ROWS_PER_BLOCK;

    for (int r = 0; r < ROWS_PER_BLOCK; ++r) {
        const long long row = row0 + r;
        const float2 xv = *(const float2*)(x + row * D_DIM + c0);

        float acc = 0.f;
        float2 yv;
        yv.x = rqs_eval(xv.x, pA, acc);
        yv.y = rqs_eval(xv.y, pB, acc);
        *(float2*)(y + row * D_DIM + c0) = yv;

        // deterministic block reduction: wave32 xor-tree, then 8 wave partials
        #pragma unroll
        for (int o = 16; o > 0; o >>= 1) acc += __shfl_xor(acc, o, 32);
        if ((tid & 31) == 0) redbuf[tid >> 5] = acc;
        __syncthreads();
        if (tid == 0) {
            float s = redbuf[0];
            #pragma unroll
            for (int wv = 1; wv < THREADS / 32; ++wv) s += redbuf[wv];
            logdet[row] = s;
        }
        __syncthreads();
    }
}

// ---------------------------------------------------------------------------
extern "C" void kernel_launch(void* const* d_in, const int* in_sizes, int n_in,
                              void* d_out, int out_size, void* d_ws, size_t ws_size,
                              hipStream_t stream) {
    const float* x  = (const float*)d_in[0];
    const float* uw = (const float*)d_in[1];
    const float* uh = (const float*)d_in[2];
    const float* ud = (const float*)d_in[3];

    float* out    = (float*)d_out;
    float* y      = out;                          // [B, D]
    float* logdet = out + (size_t)B_DIM * D_DIM;  // [B]

    float* ptab = (float*)d_ws;                   // [D, 33] knot table, 67584 bytes

    // 1) build knot tables (runs once per launch; stream-ordered before main pass)
    rqs_prep_kernel<<<(D_DIM + 255) / 256, 256, 0, stream>>>(uw, uh, ud, ptab);

    // 2) streaming spline transform + per-row logdet
    rqs_main_kernel<<<B_DIM / ROWS_PER_BLOCK, THREADS, 0, stream>>>(x, ptab, y, logdet);
}